// MultiBoxLoss_36859409335038
// MI455X (gfx1250) — compile-verified
//
#include <hip/hip_runtime.h>
#include <math.h>

#define NB 32
#define NP 16384
#define NO 16
#define NC 80
#define TILE 64
#define NBLK ((NB * NP) / TILE) /* 8192 */
#define NF4  ((TILE * NC) / 4)  /* 1280 float4 per tile */
#define ITERS (NF4 / 256)       /* 5 */

typedef __attribute__((ext_vector_type(2))) float v2f;
typedef __attribute__((ext_vector_type(8))) float v8f;
typedef __attribute__((ext_vector_type(4))) int   v4i;

typedef __attribute__((address_space(1))) v4i* as1_b128_t;
typedef __attribute__((address_space(3))) v4i* as3_b128_t;

#if __has_builtin(__builtin_amdgcn_wmma_f32_16x16x4_f32)
#define USE_WMMA_RED 1
#else
#define USE_WMMA_RED 0
#endif

#if __has_builtin(__builtin_amdgcn_global_load_async_to_lds_b128) && \
    __has_builtin(__builtin_amdgcn_s_wait_asynccnt)
#define USE_ASYNC_LDS 1
#else
#define USE_ASYNC_LDS 0
#endif

// ---------------------------------------------------------------------------
// Device helpers
// ---------------------------------------------------------------------------
__device__ __forceinline__ float balanced_l1(float diff) {
    // b = e^(R_GAMMA/R_ALPHA) - 1 = e^3 - 1
    const float bc = 19.085537f;
    float d = fabsf(diff);
    float sm = (0.5f / bc) * (bc * d + 1.0f) * log1pf(bc * d / 0.11f) - 0.5f * d;
    float bg = 1.5f * d + (1.5f / bc - 0.5f * 0.11f);
    return (d < 0.11f) ? sm : bg;
}

__device__ __forceinline__ float focal1(float x, bool t) {
    // F_ALPHA = 0.25, F_GAMMA = 1.0 (pt^1 = pt)
    float ax = fabsf(x);
    float e  = expf(-ax);
    float den = 1.0f + e;
    float p  = (x >= 0.0f) ? (1.0f / den) : (e / den); // sigmoid(x)
    float pt = t ? (1.0f - p) : p;
    float w  = (t ? 0.25f : 0.75f) * pt;
    float bce = fmaxf(x, 0.0f) - (t ? x : 0.0f) + log1pf(e);
    return bce * w;
}

#if USE_WMMA_RED
// Sum `chunks`*64 floats living in LDS array s[] using the f32 matrix core:
//   D = A(16x4) * ones(4x16) + C  =>  D[m][j] = rowsum_m(A)  (f32 FMA, exact)
// C-chained across chunks. Must be executed by one fully-active wave
// (lanes 0..31, EXEC all ones). Result valid on lane 0.
__device__ __forceinline__ float wave_wmma_sum(const float* s, int lane, int chunks) {
    const int m    = lane & 15;
    const int hi   = lane >> 4;
    const int base = (m << 2) | (hi << 1);
    v2f ones; ones.x = 1.0f; ones.y = 1.0f;
    v8f c = {0.f, 0.f, 0.f, 0.f, 0.f, 0.f, 0.f, 0.f};
#pragma unroll
    for (int ch = 0; ch < chunks; ++ch) {
        v2f a;
        a.x = s[ch * 64 + base];     // A[m][2*hi]
        a.y = s[ch * 64 + base + 1]; // A[m][2*hi+1]
        c = __builtin_amdgcn_wmma_f32_16x16x4_f32(false, a, false, ones,
                                                  (short)0, c, false, false);
    }
    // lane0 holds D[0..7][0], lane16 holds D[8..15][0]
    float part = c[0] + c[1] + c[2] + c[3] + c[4] + c[5] + c[6] + c[7];
    part += __shfl_down(part, 16);
    return part; // lane 0: total
}
#endif

// ---------------------------------------------------------------------------
// Kernel 0: init packed per-truth argmax slots.
// pack = (iou_bits << 32) | (0xFFFFFFFF - p) -> max == (max iou, lowest p)
// ---------------------------------------------------------------------------
__global__ void k_init(unsigned long long* __restrict__ gbp) {
    int i = threadIdx.x;
    if (i < NB * NO) gbp[i] = 0xFFFFFFFFull;
}

// ---------------------------------------------------------------------------
// Kernel 1: per-prior overlap/argmax over truths + per-truth argmax over priors
// grid (NP/256, NB), block 256
// ---------------------------------------------------------------------------
__global__ void k_overlap(const float* __restrict__ priors,
                          const float* __restrict__ targets,
                          float* __restrict__ ov_out,
                          int* __restrict__ idx_out,
                          unsigned long long* __restrict__ gbp) {
    __shared__ float4 s_tr[NO];
    __shared__ float  s_area[NO];
    __shared__ unsigned long long s_red[256];

    const int tid = threadIdx.x;
    const int b   = blockIdx.y;

    if (tid < NO) {
        const float* t = targets + ((size_t)b * NO + tid) * 5;
        float4 v = make_float4(t[0], t[1], t[2], t[3]);
        s_tr[tid]   = v;
        s_area[tid] = (v.z - v.x) * (v.w - v.y);
    }
    __syncthreads();

    const int p = blockIdx.x * 256 + tid;
    const float4 pr = ((const float4*)priors)[p];
    const float px1 = pr.x - pr.z * 0.5f;
    const float py1 = pr.y - pr.w * 0.5f;
    const float px2 = pr.x + pr.z * 0.5f;
    const float py2 = pr.y + pr.w * 0.5f;
    const float areap = (px2 - px1) * (py2 - py1);

    float best = -1.0f;
    int   bt   = 0;
    float ovs[NO];
#pragma unroll
    for (int t = 0; t < NO; ++t) {
        float4 tr = s_tr[t];
        float ltx = fmaxf(tr.x, px1), lty = fmaxf(tr.y, py1);
        float rbx = fminf(tr.z, px2), rby = fminf(tr.w, py2);
        float w = fmaxf(rbx - ltx, 0.0f), h = fmaxf(rby - lty, 0.0f);
        float inter = w * h;
        float iou = inter / (s_area[t] + areap - inter);
        ovs[t] = iou;
        if (iou > best) { best = iou; bt = t; } // first-max (jnp.argmax axis=0)
    }
    ov_out[(size_t)b * NP + p]  = best;
    idx_out[(size_t)b * NP + p] = bt;

    // per-truth argmax over priors (first occurrence via 0xFFFFFFFF - p)
    for (int t = 0; t < NO; ++t) {
        unsigned long long pack =
            ((unsigned long long)__float_as_uint(ovs[t]) << 32) |
            (unsigned long long)(0xFFFFFFFFu - (unsigned)p);
        s_red[tid] = pack;
        __syncthreads();
        for (int s = 128; s > 0; s >>= 1) {
            if (tid < s) {
                unsigned long long o = s_red[tid + s];
                if (o > s_red[tid]) s_red[tid] = o;
            }
            __syncthreads();
        }
        if (tid == 0) atomicMax(&gbp[b * NO + t], s_red[0]);
        __syncthreads();
    }
}

// ---------------------------------------------------------------------------
// Kernel 2: sequential override (last-wins, matches JAX scatter order)
// ---------------------------------------------------------------------------
__global__ void k_override(const unsigned long long* __restrict__ gbp,
                           float* __restrict__ ov,
                           int* __restrict__ bidx) {
    int b = threadIdx.x;
    if (b >= NB) return;
    for (int j = 0; j < NO; ++j) {
        unsigned long long pk = gbp[b * NO + j];
        unsigned p = 0xFFFFFFFFu - (unsigned)(pk & 0xFFFFFFFFull);
        if (p < NP) {
            ov[(size_t)b * NP + p]   = 2.0f;
            bidx[(size_t)b * NP + p] = j;
        }
    }
}

// ---------------------------------------------------------------------------
// Kernel 3: main loss. One block = 64 priors (64x80 logits = 20KB tile).
// Async-DMA the tile to LDS (ASYNCcnt path), compute focal from LDS,
// reduce partials with the f32 matrix core.
// ---------------------------------------------------------------------------
__global__ void k_loss(const float* __restrict__ loc,
                       const float* __restrict__ conf,
                       const float* __restrict__ priors,
                       const float* __restrict__ targets,
                       const float* __restrict__ ov,
                       const int* __restrict__ bidx,
                       float* __restrict__ part_l,
                       float* __restrict__ part_c,
                       float* __restrict__ part_n) {
    __shared__ int   s_tc[TILE];
    __shared__ float s_valid[TILE];
    __shared__ float s_red[256];
    __shared__ float s_l[TILE];
    __shared__ float s_n[TILE];
#if USE_ASYNC_LDS
    __shared__ float4 s_conf[NF4]; // 20 KB staging buffer
#endif

    const int tid = threadIdx.x;
    const size_t tile = (size_t)blockIdx.x * TILE; // flattened b*NP + p0
    const int b = (int)(tile / NP);
    const float4* cf = (const float4*)(conf + tile * NC);

#if USE_ASYNC_LDS
    // Issue the whole tile as async global->LDS b128 copies (tracked by ASYNCcnt)
#pragma unroll
    for (int it = 0; it < ITERS; ++it) {
        int i4 = it * 256 + tid;
        __builtin_amdgcn_global_load_async_to_lds_b128(
            (as1_b128_t)(cf + i4),
            (as3_b128_t)(&s_conf[i4]),
            0, 0);
    }
#endif

    float loc_sum = 0.0f, cnt = 0.0f;
    if (tid < TILE) {
        size_t gp = tile + tid;
        int pp = (int)(gp % NP);
        float o = ov[gp];
        int   j = bidx[gp];
        bool pos = (o >= 0.5f);
        bool neg = (o < 0.4f);
        const float* tr = targets + ((size_t)b * NO + j) * 5;
        s_tc[tid]    = pos ? ((int)tr[4] + 1) : 0;
        s_valid[tid] = (pos || neg) ? 1.0f : 0.0f;
        cnt = pos ? 1.0f : 0.0f;
        if (pos) {
            float mx1 = tr[0], my1 = tr[1], mx2 = tr[2], my2 = tr[3];
            float4 pr = ((const float4*)priors)[pp];
            float ecx = ((mx1 + mx2) * 0.5f - pr.x) / (0.1f * pr.z);
            float ecy = ((my1 + my2) * 0.5f - pr.y) / (0.1f * pr.w);
            float ew  = logf((mx2 - mx1) / pr.z) / 0.2f;
            float eh  = logf((my2 - my1) / pr.w) / 0.2f;
            float4 ld = ((const float4*)loc)[gp];
            loc_sum = balanced_l1(ld.x - ecx) + balanced_l1(ld.y - ecy) +
                      balanced_l1(ld.z - ew)  + balanced_l1(ld.w - eh);
        }
    }
    __syncthreads();

#if USE_ASYNC_LDS
    // Each thread consumes exactly the addresses it issued -> per-wave wait.
    __builtin_amdgcn_s_wait_asynccnt(0);
    asm volatile("" ::: "memory");
#endif

    float csum = 0.0f;
#pragma unroll
    for (int it = 0; it < ITERS; ++it) {
        int i4 = it * 256 + tid;
#if USE_ASYNC_LDS
        float4 x = s_conf[i4];                    // ds_load_b128
#else
        __builtin_prefetch(&cf[i4 + 256], 0, 0);  // global_prefetch_b8
        float4 x = cf[i4];                        // global_load_b128
#endif
        int li = i4 / (NC / 4);
        int c0 = (i4 % (NC / 4)) * 4;
        float v  = s_valid[li];
        int   tc = s_tc[li];
        float f = focal1(x.x, tc == c0 + 1) + focal1(x.y, tc == c0 + 2) +
                  focal1(x.z, tc == c0 + 3) + focal1(x.w, tc == c0 + 4);
        csum += v * f;
    }

    // Block reductions
    s_red[tid] = csum;
    if (tid < TILE) { s_l[tid] = loc_sum; s_n[tid] = cnt; }
    __syncthreads();

#if USE_WMMA_RED
    if (tid < 32) { // wave 0 fully active: EXEC all ones as WMMA requires
        float tot_c = wave_wmma_sum(s_red, tid, 4);
        float tot_l = wave_wmma_sum(s_l, tid, 1);
        float tot_n = wave_wmma_sum(s_n, tid, 1);
        if (tid == 0) {
            part_c[blockIdx.x] = tot_c;
            part_l[blockIdx.x] = tot_l;
            part_n[blockIdx.x] = tot_n;
        }
    }
#else
    for (int s = 128; s > 0; s >>= 1) {
        if (tid < s) s_red[tid] += s_red[tid + s];
        __syncthreads();
    }
    if (tid == 0) part_c[blockIdx.x] = s_red[0];
    __syncthreads();
    s_red[tid] = (tid < TILE) ? loc_sum : 0.0f;
    __syncthreads();
    for (int s = 128; s > 0; s >>= 1) {
        if (tid < s) s_red[tid] += s_red[tid + s];
        __syncthreads();
    }
    if (tid == 0) part_l[blockIdx.x] = s_red[0];
    __syncthreads();
    s_red[tid] = (tid < TILE) ? cnt : 0.0f;
    __syncthreads();
    for (int s = 128; s > 0; s >>= 1) {
        if (tid < s) s_red[tid] += s_red[tid + s];
        __syncthreads();
    }
    if (tid == 0) part_n[blockIdx.x] = s_red[0];
#endif
}

// ---------------------------------------------------------------------------
// Kernel 4: final deterministic reduction in double, write two scalars
// ---------------------------------------------------------------------------
__global__ void k_final(const float* __restrict__ part_l,
                        const float* __restrict__ part_c,
                        const float* __restrict__ part_n,
                        float* __restrict__ out) {
    __shared__ double sd[256];
    __shared__ double res[3];
    const int tid = threadIdx.x;
    double sl = 0.0, sc = 0.0, sn = 0.0;
    for (int i = tid; i < NBLK; i += 256) {
        sl += (double)part_l[i];
        sc += (double)part_c[i];
        sn += (double)part_n[i];
    }
    sd[tid] = sl;
    __syncthreads();
    for (int s = 128; s > 0; s >>= 1) { if (tid < s) sd[tid] += sd[tid + s]; __syncthreads(); }
    if (tid == 0) res[0] = sd[0];
    __syncthreads();
    sd[tid] = sc;
    __syncthreads();
    for (int s = 128; s > 0; s >>= 1) { if (tid < s) sd[tid] += sd[tid + s]; __syncthreads(); }
    if (tid == 0) res[1] = sd[0];
    __syncthreads();
    sd[tid] = sn;
    __syncthreads();
    for (int s = 128; s > 0; s >>= 1) { if (tid < s) sd[tid] += sd[tid + s]; __syncthreads(); }
    if (tid == 0) {
        res[2] = sd[0];
        double pn = res[2] > 1.0 ? res[2] : 1.0;
        out[0] = (float)(res[0] / (pn * 4.0));
        out[1] = (float)(res[1] / pn);
    }
}

// ---------------------------------------------------------------------------
// Host launcher
// ---------------------------------------------------------------------------
extern "C" void kernel_launch(void* const* d_in, const int* in_sizes, int n_in,
                              void* d_out, int out_size, void* d_ws, size_t ws_size,
                              hipStream_t stream) {
    (void)in_sizes; (void)n_in; (void)out_size; (void)ws_size;
    const float* loc     = (const float*)d_in[0]; // (B,P,4)
    const float* conf    = (const float*)d_in[1]; // (B,P,80)
    const float* priors  = (const float*)d_in[2]; // (P,4)
    const float* targets = (const float*)d_in[3]; // (B,O,5)
    float* out = (float*)d_out;

    char* ws = (char*)d_ws;
    float* ov   = (float*)ws;                                       // B*P floats
    int*   bidx = (int*)(ws + 4ull * NB * NP);                      // B*P ints
    unsigned long long* gbp =
        (unsigned long long*)(ws + 8ull * NB * NP);                 // B*O u64
    float* part_l = (float*)(ws + 8ull * NB * NP + 8ull * NB * NO); // NBLK
    float* part_c = part_l + NBLK;
    float* part_n = part_c + NBLK;

    k_init<<<1, 512, 0, stream>>>(gbp);
    k_overlap<<<dim3(NP / 256, NB), 256, 0, stream>>>(priors, targets, ov, bidx, gbp);
    k_override<<<1, 32, 0, stream>>>(gbp, ov, bidx);
    k_loss<<<NBLK, 256, 0, stream>>>(loc, conf, priors, targets, ov, bidx,
                                     part_l, part_c, part_n);
    k_final<<<1, 256, 0, stream>>>(part_l, part_c, part_n, out);
}